// MeanAggregator_42502996361303
// MI455X (gfx1250) — compile-verified
//
#include <hip/hip_runtime.h>

// MeanAggregator: out[n] = (cnt==0) ? F[nodes[n]]
//                                   : (1/cnt) * sum_{s<cnt} F[neigh_idx[n,s]]
// N=100000, S=10, D=128, fp32. Memory-bound sparse gather; the 51 MB feature
// table is L2-resident on MI455X (192 MB L2). One wave32 per node: one 512 B
// feature row == 32 lanes x global_load_b128 (saddr form). Node metadata is
// wave-uniform (readfirstlane) -> SMEM s_load; next node's indices are
// software-pipelined one iteration ahead so its feature rows can be
// prefetched (global_prefetch_b8) while the current node accumulates.

typedef float v4f __attribute__((ext_vector_type(4)));

// SS > 0 and DD > 0: compile-time S and D (fully specialized fast path,
// DD must be 128 so one lane == one float4 chunk). SS == 0: runtime fallback.
template <int SS, int DD>
__global__ __launch_bounds__(256) void mean_agg_kernel(
    const int* __restrict__ nodes,
    const int* __restrict__ neigh_idx,
    const int* __restrict__ neigh_count,
    const float* __restrict__ features,
    float* __restrict__ out,
    int N, int Sdyn, int Ddyn)
{
    const int S    = (SS > 0) ? SS : Sdyn;
    const int D    = (DD > 0) ? DD : Ddyn;
    const int D4   = D >> 2;
    const int lane = (int)(threadIdx.x & 31u);
    const int wave = (int)((blockIdx.x * blockDim.x + threadIdx.x) >> 5);
    const int nwav = (int)((gridDim.x * blockDim.x) >> 5);

    const v4f* __restrict__ feat4 = (const v4f*)features;
    v4f* __restrict__ out4        = (v4f*)out;

    // Provably wave-uniform node index -> metadata loads lower to SMEM and
    // gather row bases are SALU-computed (saddr + lane voffset).
    int n = __builtin_amdgcn_readfirstlane(wave);

    if constexpr (SS > 0 && DD > 0) {
        // ---------- specialized fast path (S=SS, D=DD=128) ----------
        if (n >= N) return;

        // Pipeline stage 0: metadata for the first node (uniform s_loads).
        int cnt = neigh_count[n];
        int idx[SS];
#pragma unroll
        for (int s = 0; s < SS; ++s) idx[s] = neigh_idx[(size_t)n * SS + s];

        for (; n < N; n += nwav) {
            // Load NEXT node's metadata early and prefetch its gather rows
            // (covers the cold/HBM pass; warm L2 hits make these no-ops).
            const int n2 = n + nwav;
            int cnt2 = 0;
            int idx2[SS];
#pragma unroll
            for (int s = 0; s < SS; ++s) idx2[s] = 0;
            if (n2 < N) {
                cnt2 = neigh_count[n2];
#pragma unroll
                for (int s = 0; s < SS; ++s) idx2[s] = neigh_idx[(size_t)n2 * SS + s];
#pragma unroll
                for (int s = 0; s < SS; ++s) {
                    const float* row = features + (size_t)idx2[s] * DD;
                    __builtin_prefetch(row, 0, 0);         // global_prefetch_b8
                    __builtin_prefetch(row + 64, 0, 0);    // +256 B: rest of 512 B row
                }
            }

            // ---- current node: D4 == 32 == lanes, no chunk loop ----
            const size_t outBase = (size_t)n * (DD / 4) + lane;
            if (cnt == 0) {
                // Self-feature fallback (wave-uniform branch).
                const int self = nodes[n];                 // uniform -> s_load
                v4f v = feat4[(size_t)self * (DD / 4) + lane];
                __builtin_nontemporal_store(v, &out4[outBase]);
            } else {
                const float inv = 1.0f / (float)cnt;
                v4f acc = (v4f)0.0f;
                // Branchless over all SS slots (padded indices are valid in
                // [0,N)): SS independent b128 gathers -> one load clause.
#pragma unroll
                for (int s = 0; s < SS; ++s) {
                    const float w = (s < cnt) ? inv : 0.0f;
                    v4f v = feat4[(size_t)idx[s] * (DD / 4) + lane];
                    acc += w * v;
                }
                // Write-once output: NT store keeps the table resident in L2.
                __builtin_nontemporal_store(acc, &out4[outBase]);
            }

            // Rotate pipeline.
            cnt = cnt2;
#pragma unroll
            for (int s = 0; s < SS; ++s) idx[s] = idx2[s];
        }
    } else {
        // ---------- generic fallback (runtime S/D, D % 4 == 0) ----------
        for (; n < N; n += nwav) {
            const int cnt = neigh_count[n];
            const int* __restrict__ ip = neigh_idx + (size_t)n * S;

            const int n2 = n + nwav;
            if (n2 < N) {
                __builtin_prefetch(neigh_count + n2, 0, 0);
                __builtin_prefetch(neigh_idx + (size_t)n2 * S, 0, 0);
            }

            if (cnt == 0) {
                const int self = nodes[n];
                for (int c = lane; c < D4; c += 32) {
                    v4f v = feat4[(size_t)self * D4 + c];
                    __builtin_nontemporal_store(v, &out4[(size_t)n * D4 + c]);
                }
            } else {
                const float inv = 1.0f / (float)cnt;
                for (int c = lane; c < D4; c += 32) {
                    v4f acc = (v4f)0.0f;
                    for (int s = 0; s < S; ++s) {
                        const int   id = ip[s];            // uniform load
                        const float w  = (s < cnt) ? inv : 0.0f;
                        v4f v = feat4[(size_t)id * D4 + c];
                        acc += w * v;
                    }
                    __builtin_nontemporal_store(acc, &out4[(size_t)n * D4 + c]);
                }
            }
        }
    }
}

extern "C" void kernel_launch(void* const* d_in, const int* in_sizes, int n_in,
                              void* d_out, int out_size, void* d_ws, size_t ws_size,
                              hipStream_t stream)
{
    // setup_inputs() order: nodes[N], neigh_idx[N*S], neigh_count[N], features[N*D]
    const int*   nodes       = (const int*)d_in[0];
    const int*   neigh_idx   = (const int*)d_in[1];
    const int*   neigh_count = (const int*)d_in[2];
    const float* features    = (const float*)d_in[3];
    float*       out         = (float*)d_out;

    const int N = in_sizes[2];
    const int S = (N > 0) ? in_sizes[1] / N : 0;
    const int D = (N > 0) ? in_sizes[3] / N : 0;

    if (N <= 0 || D <= 0) return;

    // 256 threads = 8 wave32 per block; ~4 nodes per wave via grid-stride so
    // the prefetch pipeline has a "next" iteration.
    const int wavesPerBlock = 256 / 32;
    const int nodesPerWave  = 4;
    int blocks = (N + nodesPerWave * wavesPerBlock - 1) / (nodesPerWave * wavesPerBlock);
    if (blocks < 1) blocks = 1;

    if (S == 10 && D == 128) {
        mean_agg_kernel<10, 128><<<blocks, 256, 0, stream>>>(
            nodes, neigh_idx, neigh_count, features, out, N, S, D);
    } else {
        mean_agg_kernel<0, 0><<<blocks, 256, 0, stream>>>(
            nodes, neigh_idx, neigh_count, features, out, N, S, D);
    }
}